// KBGN_78391743087163
// MI455X (gfx1250) — compile-verified
//
#include <hip/hip_runtime.h>
#include <hip/hip_bf16.h>
#include <cstddef>

// ---------------------------------------------------------------------------
// Shapes (from reference): V=11322, EMB=300, ELMO=1024, WORD=512, H=512,
// B=32, R=10, L=20.  N = B*R = 320 sequences per encoder.
// LSTM layer-0 input size IN0 = 300 + 512 = 812 (multiple of 4 -> clean K loop).
// ---------------------------------------------------------------------------

#define NSEQ   320
#define HDIM   512
#define GDIM   2048          // 4*H
#define IN0    812
#define EMBD   300
#define ELMOD  1024
#define TQ     20
#define TH2    40
#define BB     32
#define RR     10

typedef __attribute__((ext_vector_type(2))) float v2f;
typedef __attribute__((ext_vector_type(4))) float v4f;
typedef __attribute__((ext_vector_type(8))) float v8f;

union V8 { v8f v; float f[8]; };

__device__ __forceinline__ v8f wmma_f32_k4(v2f a, v2f b, v8f c) {
    // D = A(16x4,f32) * B(4x16,f32) + C(16x16,f32)  -> v_wmma_f32_16x16x4_f32
    return __builtin_amdgcn_wmma_f32_16x16x4_f32(
        /*neg_a=*/false, a, /*neg_b=*/false, b,
        /*c_mod=*/(short)0, c, /*reuse_a=*/false, /*reuse_b=*/false);
}

__device__ __forceinline__ float sigf(float x) { return 1.0f / (1.0f + __expf(-x)); }

// ---------------------------------------------------------------------------
// zero fill
// ---------------------------------------------------------------------------
__global__ void zero_kernel(float* __restrict__ p, int n) {
    int i = blockIdx.x * blockDim.x + threadIdx.x;
    if (i < n) p[i] = 0.0f;
}

// ---------------------------------------------------------------------------
// x[row, 0:300] = w_emb[tok[row], :]      (row = 0..Mtot-1 over (n,t))
// ---------------------------------------------------------------------------
__global__ void gather_emb_kernel(const int* __restrict__ tok,
                                  const float* __restrict__ w_emb,
                                  float* __restrict__ x, int Mtot) {
    int idx = blockIdx.x * blockDim.x + threadIdx.x;
    int total = Mtot * EMBD;
    if (idx >= total) return;
    int row = idx / EMBD;
    int e   = idx - row * EMBD;
    x[(size_t)row * IN0 + e] = w_emb[(size_t)tok[row] * EMBD + e];
}

// ---------------------------------------------------------------------------
// ELMo projection: x[row, 300 + g] = sum_k elmo[tok[row],k] * Wc[g,k] + bc[g]
// One wave per 16x64 strip of C[Mtot x 512]: 4 accumulators share each
// A-fragment load (1.25 loads / wmma instead of 2).  K = 1024 in chunks of 4.
// ---------------------------------------------------------------------------
__global__ void elmo_proj_wmma(const int* __restrict__ tok,
                               const float* __restrict__ elmo,
                               const float* __restrict__ Wc,
                               const float* __restrict__ bc,
                               float* __restrict__ x, int Mtot) {
    const int NG = HDIM / 64;                         // 8 column groups of 4 tiles
    int wave = blockIdx.x * (blockDim.x >> 5) + (threadIdx.x >> 5);
    int lane = threadIdx.x & 31;
    int totalW = (Mtot >> 4) * NG;
    if (wave >= totalW) return;
    int mt = wave / NG, ng = wave - mt * NG;
    int rl = lane & 15;
    int row = mt * 16 + rl;                           // A row for this lane
    int koff = (lane < 16) ? 0 : 2;                   // upper half-wave holds K+2/K+3
    const float* ap = elmo + (size_t)tok[row] * ELMOD + koff;
    const float* bp0 = Wc + (size_t)(ng * 64 +  0 + rl) * ELMOD + koff;
    const float* bp1 = Wc + (size_t)(ng * 64 + 16 + rl) * ELMOD + koff;
    const float* bp2 = Wc + (size_t)(ng * 64 + 32 + rl) * ELMOD + koff;
    const float* bp3 = Wc + (size_t)(ng * 64 + 48 + rl) * ELMOD + koff;
    v8f acc0 = {}, acc1 = {}, acc2 = {}, acc3 = {};
    #pragma unroll 2
    for (int k = 0; k < ELMOD; k += 4) {
        v2f a  = { ap[k],  ap[k + 1]  };
        v2f b0 = { bp0[k], bp0[k + 1] };
        v2f b1 = { bp1[k], bp1[k + 1] };
        v2f b2 = { bp2[k], bp2[k + 1] };
        v2f b3 = { bp3[k], bp3[k + 1] };
        acc0 = wmma_f32_k4(a, b0, acc0);
        acc1 = wmma_f32_k4(a, b1, acc1);
        acc2 = wmma_f32_k4(a, b2, acc2);
        acc3 = wmma_f32_k4(a, b3, acc3);
    }
    int rbase = mt * 16 + ((lane < 16) ? 0 : 8);
    V8 u0, u1, u2, u3;
    u0.v = acc0; u1.v = acc1; u2.v = acc2; u3.v = acc3;
    float bv0 = bc[ng * 64 +  0 + rl];
    float bv1 = bc[ng * 64 + 16 + rl];
    float bv2 = bc[ng * 64 + 32 + rl];
    float bv3 = bc[ng * 64 + 48 + rl];
    #pragma unroll
    for (int v = 0; v < 8; ++v) {
        float* xr = x + (size_t)(rbase + v) * IN0 + EMBD + ng * 64 + rl;
        xr[ 0] = u0.f[v] + bv0;
        xr[16] = u1.f[v] + bv1;
        xr[32] = u2.f[v] + bv2;
        xr[48] = u3.f[v] + bv3;
    }
}

// ---------------------------------------------------------------------------
// Fused per-timestep gate GEMM:
//   gates[320 x 2048] = bias + X_t[320 x K1] @ Wih^T + Hprev[320 x 512] @ Whh^T
// One wave per 16x64 strip (4 C tiles, shared A fragments); 20 x 32 = 640 waves.
// ---------------------------------------------------------------------------
__global__ void lstm_gate_wmma(const float* __restrict__ X, int ldx, int xoff, int K1,
                               const float* __restrict__ Wih,
                               const float* __restrict__ Hprev,
                               const float* __restrict__ Whh,
                               const float* __restrict__ bias,
                               float* __restrict__ gates) {
    const int NG = GDIM / 64;                         // 32 column groups
    const int MT = NSEQ / 16;                         // 20
    int wave = blockIdx.x * (blockDim.x >> 5) + (threadIdx.x >> 5);
    int lane = threadIdx.x & 31;
    if (wave >= MT * NG) return;
    int mt = wave / NG, ng = wave - mt * NG;
    int rl = lane & 15;
    int row = mt * 16 + rl;
    int koff = (lane < 16) ? 0 : 2;
    int c0 = ng * 64 + rl;

    v8f acc0 = {}, acc1 = {}, acc2 = {}, acc3 = {};
    {   // input contribution: X_t @ Wih^T
        const float* ap  = X + (size_t)row * ldx + xoff + koff;
        const float* bp0 = Wih + (size_t)(c0 +  0) * K1 + koff;
        const float* bp1 = Wih + (size_t)(c0 + 16) * K1 + koff;
        const float* bp2 = Wih + (size_t)(c0 + 32) * K1 + koff;
        const float* bp3 = Wih + (size_t)(c0 + 48) * K1 + koff;
        #pragma unroll 2
        for (int k = 0; k < K1; k += 4) {
            v2f a  = { ap[k],  ap[k + 1]  };
            v2f b0 = { bp0[k], bp0[k + 1] };
            v2f b1 = { bp1[k], bp1[k + 1] };
            v2f b2 = { bp2[k], bp2[k + 1] };
            v2f b3 = { bp3[k], bp3[k + 1] };
            acc0 = wmma_f32_k4(a, b0, acc0);
            acc1 = wmma_f32_k4(a, b1, acc1);
            acc2 = wmma_f32_k4(a, b2, acc2);
            acc3 = wmma_f32_k4(a, b3, acc3);
        }
    }
    {   // recurrent contribution: Hprev @ Whh^T
        const float* ap  = Hprev + (size_t)row * HDIM + koff;
        const float* bp0 = Whh + (size_t)(c0 +  0) * HDIM + koff;
        const float* bp1 = Whh + (size_t)(c0 + 16) * HDIM + koff;
        const float* bp2 = Whh + (size_t)(c0 + 32) * HDIM + koff;
        const float* bp3 = Whh + (size_t)(c0 + 48) * HDIM + koff;
        #pragma unroll 2
        for (int k = 0; k < HDIM; k += 4) {
            v2f a  = { ap[k],  ap[k + 1]  };
            v2f b0 = { bp0[k], bp0[k + 1] };
            v2f b1 = { bp1[k], bp1[k + 1] };
            v2f b2 = { bp2[k], bp2[k + 1] };
            v2f b3 = { bp3[k], bp3[k + 1] };
            acc0 = wmma_f32_k4(a, b0, acc0);
            acc1 = wmma_f32_k4(a, b1, acc1);
            acc2 = wmma_f32_k4(a, b2, acc2);
            acc3 = wmma_f32_k4(a, b3, acc3);
        }
    }
    int rbase = mt * 16 + ((lane < 16) ? 0 : 8);
    V8 u0, u1, u2, u3;
    u0.v = acc0; u1.v = acc1; u2.v = acc2; u3.v = acc3;
    float bv0 = bias[c0 +  0];
    float bv1 = bias[c0 + 16];
    float bv2 = bias[c0 + 32];
    float bv3 = bias[c0 + 48];
    #pragma unroll
    for (int v = 0; v < 8; ++v) {
        float* gr = gates + (size_t)(rbase + v) * GDIM + c0;
        gr[ 0] = u0.f[v] + bv0;
        gr[16] = u1.f[v] + bv1;
        gr[32] = u2.f[v] + bv2;
        gr[48] = u3.f[v] + bv3;
    }
}

// ---------------------------------------------------------------------------
// LSTM cell pointwise (PyTorch gate order i,f,g,o):
//   c = sig(f)*c + sig(i)*tanh(g) ;  h = sig(o)*tanh(c)
// Optionally stores h into the full sequence buffer (layer-0 output) and/or
// scatters h into outsel when t == len[n]-1 (DynamicRNN last-valid hidden).
// ---------------------------------------------------------------------------
__global__ void lstm_pointwise(const float* __restrict__ gates,
                               float* __restrict__ h, float* __restrict__ c,
                               float* __restrict__ hall, int T, int t,
                               float* __restrict__ outsel,
                               const int* __restrict__ lens) {
    int idx = blockIdx.x * blockDim.x + threadIdx.x;
    if (idx >= NSEQ * HDIM) return;
    int n  = idx >> 9;          // /512
    int hh = idx & (HDIM - 1);
    const float* g = gates + (size_t)n * GDIM;
    float ig = g[hh];
    float fg = g[HDIM + hh];
    float gg = g[2 * HDIM + hh];
    float og = g[3 * HDIM + hh];
    float cn = sigf(fg) * c[idx] + sigf(ig) * tanhf(gg);
    float hn = sigf(og) * tanhf(cn);
    c[idx] = cn;
    h[idx] = hn;
    if (hall)   hall[((size_t)n * T + t) * HDIM + hh] = hn;
    if (outsel) { if (t == lens[n] - 1) outsel[idx] = hn; }
}

// ---------------------------------------------------------------------------
// nz[r] = (sum_c |hist_embed[r,c]| != 0)   — matches t_rel.abs().sum(-1).bool()
// ---------------------------------------------------------------------------
__global__ void nz_kernel(const float* __restrict__ he, int* __restrict__ nz) {
    int r = blockIdx.x * blockDim.x + threadIdx.x;
    if (r >= NSEQ) return;
    float s = 0.0f;
    const float* p = he + (size_t)r * HDIM;
    for (int cI = 0; cI < HDIM; ++cI) s += fabsf(p[cI]);
    nz[r] = (s != 0.0f) ? 1 : 0;
}

// ---------------------------------------------------------------------------
// text_rel[b,i,j,k, 0:512 ]  = hist[b,j] * [j<=i] * [k<=i && nz(b,k)]
// text_rel[b,i,j,k, 512:1024] = hist[b,k] * [k<=i] * [j<=i && nz(b,j)]
// float4-vectorized 131 MB stream write; stores use the NT temporal hint so
// this write-once data does not evict the LSTM working set from L2.
// ---------------------------------------------------------------------------
__global__ void text_rel_kernel(const float* __restrict__ he,
                                const int* __restrict__ nz,
                                float* __restrict__ out) {
    long long idx = (long long)blockIdx.x * blockDim.x + threadIdx.x;
    const long long total = (long long)BB * RR * RR * RR * (2 * HDIM / 4); // 8,192,000
    if (idx >= total) return;
    int c4 = (int)(idx & 255);        // 1024/4 = 256 float4 per (b,i,j,k)
    long long rem = idx >> 8;
    int k = (int)(rem % RR); rem /= RR;
    int j = (int)(rem % RR); rem /= RR;
    int i = (int)(rem % RR);
    int b = (int)(rem / RR);
    int c = c4 * 4;
    v4f val = {};
    if (c < HDIM) {
        if ((j <= i) && (k <= i) && nz[b * RR + k]) {
            val = *(const v4f*)(he + ((size_t)(b * RR + j)) * HDIM + c);
        }
    } else {
        if ((k <= i) && (j <= i) && nz[b * RR + j]) {
            val = *(const v4f*)(he + ((size_t)(b * RR + k)) * HDIM + (c - HDIM));
        }
    }
    __builtin_nontemporal_store(val, ((v4f*)out) + idx);
}

// ---------------------------------------------------------------------------
// Host-side encoder driver (launch sequence only; all on `stream`).
// ---------------------------------------------------------------------------
static void run_encode(const int* tok, const int* lens, int T,
                       const float* w_emb, const float* elmo_emb,
                       const float* Wc, const float* bc,
                       const float* Wih0, const float* Whh0, const float* b0,
                       const float* Wih1, const float* Whh1, const float* b1,
                       float* xbuf, float* h1all, float* gatesbuf,
                       float* hstate, float* cstate,
                       float* outsel, hipStream_t stream) {
    const int Mtot = NSEQ * T;

    // x = [emb | elmoProj]
    {
        int tot = Mtot * EMBD;
        gather_emb_kernel<<<(tot + 255) / 256, 256, 0, stream>>>(tok, w_emb, xbuf, Mtot);
    }
    {
        int wv = (Mtot / 16) * (HDIM / 64);
        elmo_proj_wmma<<<(wv + 7) / 8, 256, 0, stream>>>(tok, elmo_emb, Wc, bc, xbuf, Mtot);
    }

    const int gateBlocks = ((NSEQ / 16) * (GDIM / 64) + 7) / 8;   // 640 waves / 8
    const int pwBlocks   = (NSEQ * HDIM + 255) / 256;             // 640

    // layer 0
    zero_kernel<<<(2 * NSEQ * HDIM + 255) / 256, 256, 0, stream>>>(hstate, 2 * NSEQ * HDIM);
    for (int t = 0; t < T; ++t) {
        lstm_gate_wmma<<<gateBlocks, 256, 0, stream>>>(xbuf, T * IN0, t * IN0, IN0,
                                                       Wih0, hstate, Whh0, b0, gatesbuf);
        lstm_pointwise<<<pwBlocks, 256, 0, stream>>>(gatesbuf, hstate, cstate,
                                                     h1all, T, t, nullptr, nullptr);
    }
    // layer 1
    zero_kernel<<<(2 * NSEQ * HDIM + 255) / 256, 256, 0, stream>>>(hstate, 2 * NSEQ * HDIM);
    for (int t = 0; t < T; ++t) {
        lstm_gate_wmma<<<gateBlocks, 256, 0, stream>>>(h1all, T * HDIM, t * HDIM, HDIM,
                                                       Wih1, hstate, Whh1, b1, gatesbuf);
        lstm_pointwise<<<pwBlocks, 256, 0, stream>>>(gatesbuf, hstate, cstate,
                                                     nullptr, T, t, outsel, lens);
    }
}

extern "C" void kernel_launch(void* const* d_in, const int* in_sizes, int n_in,
                              void* d_out, int out_size, void* d_ws, size_t ws_size,
                              hipStream_t stream) {
    (void)in_sizes; (void)n_in; (void)out_size; (void)ws_size;

    const int*   ques      = (const int*)  d_in[0];
    const int*   hist      = (const int*)  d_in[1];
    const int*   ques_len  = (const int*)  d_in[2];
    const int*   hist_len  = (const int*)  d_in[3];
    // d_in[4] img_feat, d_in[5] v_relations: unused by the reference output
    const float* w_emb     = (const float*)d_in[6];
    const float* elmo_emb  = (const float*)d_in[7];
    const float* Wc        = (const float*)d_in[8];
    const float* bc        = (const float*)d_in[9];
    const float* q_Wih0 = (const float*)d_in[10];
    const float* q_Whh0 = (const float*)d_in[11];
    const float* q_b0   = (const float*)d_in[12];
    const float* q_Wih1 = (const float*)d_in[13];
    const float* q_Whh1 = (const float*)d_in[14];
    const float* q_b1   = (const float*)d_in[15];
    const float* h_Wih0 = (const float*)d_in[16];
    const float* h_Whh0 = (const float*)d_in[17];
    const float* h_b0   = (const float*)d_in[18];
    const float* h_Wih1 = (const float*)d_in[19];
    const float* h_Whh1 = (const float*)d_in[20];
    const float* h_b1   = (const float*)d_in[21];

    float* out = (float*)d_out;
    float* ques_embed = out;                              // [B,R,H]  = 163,840 floats
    float* text_rel   = out + (size_t)BB * RR * HDIM;     // [B,R,R,R,2H]

    // Workspace layout (floats)
    float* w = (float*)d_ws;
    const size_t OFF_X     = 0;                                        // 320*40*812
    const size_t OFF_H1    = OFF_X   + (size_t)NSEQ * TH2 * IN0;       // 320*40*512
    const size_t OFF_GATES = OFF_H1  + (size_t)NSEQ * TH2 * HDIM;      // 320*2048
    const size_t OFF_HST   = OFF_GATES + (size_t)NSEQ * GDIM;          // 320*512
    const size_t OFF_CST   = OFF_HST + (size_t)NSEQ * HDIM;            // 320*512
    const size_t OFF_HE    = OFF_CST + (size_t)NSEQ * HDIM;            // 320*512 hist_embed
    const size_t OFF_NZ    = OFF_HE  + (size_t)NSEQ * HDIM;            // 320 ints

    float* xbuf     = w + OFF_X;
    float* h1all    = w + OFF_H1;
    float* gatesbuf = w + OFF_GATES;
    float* hstate   = w + OFF_HST;      // cstate directly follows (zeroed together)
    float* cstate   = w + OFF_CST;
    float* histemb  = w + OFF_HE;
    int*   nz       = (int*)(w + OFF_NZ);

    // Question encoder -> ques_embed (written straight into d_out)
    run_encode(ques, ques_len, TQ, w_emb, elmo_emb, Wc, bc,
               q_Wih0, q_Whh0, q_b0, q_Wih1, q_Whh1, q_b1,
               xbuf, h1all, gatesbuf, hstate, cstate, ques_embed, stream);

    // History encoder -> hist_embed (workspace)
    run_encode(hist, hist_len, TH2, w_emb, elmo_emb, Wc, bc,
               h_Wih0, h_Whh0, h_b0, h_Wih1, h_Whh1, h_b1,
               xbuf, h1all, gatesbuf, hstate, cstate, histemb, stream);

    // Masks + text_rel expansion
    nz_kernel<<<(NSEQ + 255) / 256, 256, 0, stream>>>(histemb, nz);
    {
        long long tot4 = (long long)BB * RR * RR * RR * (2 * HDIM / 4); // 8,192,000
        int blocks = (int)((tot4 + 255) / 256);
        text_rel_kernel<<<blocks, 256, 0, stream>>>(histemb, nz, text_rel);
    }
}